// WaveCell_21887153340817
// MI455X (gfx1250) — compile-verified
//
#include <hip/hip_runtime.h>
#include <cstdint>

// Problem constants (from reference setup_inputs / module constants)
#define BATCH 8
#define NY 1024
#define NX 1024
// dt = 0.5 -> dt^-1 = 2, dt^-2 = 4 ; h = 1 ; b0 = uth = 1 ; c_nl = 0.01

// Tiling: one 128x32 tile of one batch plane per 256-thread block (8 wave32 waves)
#define TILE_X 128
#define TILE_Y 32
#define PITCH  136               // 4-float left pad + 128 + 4-float right pad (16B chunks)
#define LROWS  (TILE_Y + 2)      // +1 halo row top/bottom
#define NCHUNK (PITCH / 4)       // 34 float4 chunks per LDS row

typedef float v4f __attribute__((ext_vector_type(4)));

__device__ __forceinline__ int clampi(int v, int lo, int hi) {
    return v < lo ? lo : (v > hi ? hi : v);
}

__global__ __launch_bounds__(256)
void wave_fdtd_kernel(const float* __restrict__ h1,
                      const float* __restrict__ h2,
                      const float* __restrict__ cl,
                      const float* __restrict__ rho,
                      const float* __restrict__ bg,
                      float* __restrict__ out)
{
    __shared__ __align__(16) float tile[LROWS * PITCH];

    const int tid = threadIdx.x;
    const int tx  = tid & 31;        // lane within wave32
    const int ty  = tid >> 5;        // wave id 0..7
    const int x0  = blockIdx.x * TILE_X;
    const int y0  = blockIdx.y * TILE_Y;
    const int b   = blockIdx.z;
    const size_t planeBase = (size_t)b * NY * NX;

    // Low 32 bits of the flat shared-aperture address == LDS byte offset.
    const uint32_t ldsBase = (uint32_t)(uintptr_t)tile;

    // ---- Stage h1 tile (+halo ring) into LDS via gfx1250 async global->LDS
    //      copies (ASYNCcnt path). Wave-row x lane-chunk decomposition keeps
    //      address gen to shifts/adds (no integer division). ----
    for (int row = ty; row < LROWS; row += 8) {
        const int gy = clampi(y0 - 1 + row, 0, NY - 1);   // clamp addresses in-bounds;
        const float* rowp = h1 + planeBase + (size_t)gy * NX;  // zero-pad applied at compute
        const uint32_t ldsRow = ldsBase + (uint32_t)(row * PITCH) * 4u;
        for (int ch = tx; ch < NCHUNK; ch += 32) {
            const int gx = clampi(x0 - 4 + ch * 4, 0, NX - 4);
            const float* src = rowp + gx;                  // 16B aligned
            const uint32_t dst = ldsRow + (uint32_t)(ch * 4) * 4u;
            asm volatile("global_load_async_to_lds_b128 %0, %1, off"
                         :: "v"(dst), "v"(src)
                         : "memory");
        }
    }
    // Each wave drains its own async queue; the barrier then makes all LDS
    // writes visible to every wave.
    asm volatile("s_wait_asynccnt 0x0" ::: "memory");
    __syncthreads();

    // ---- Compute: 256 threads x 4 iterations x float4 = 128x32 tile ----
    const int lx = tx * 4;          // tile-local column of first element
    const int gx = x0 + lx;
    const bool lftOk = (gx > 0);
    const bool rgtOk = (gx + 4 < NX);
    const size_t TOT = (size_t)BATCH * NY * NX;

#pragma unroll
    for (int r = 0; r < 4; ++r) {
        const int ly   = ty + r * 8;         // 0..31
        const int gy   = y0 + ly;
        const int lrow = ly + 1;             // halo offset
        const float* crow = &tile[lrow * PITCH + 4 + lx];

        const v4f c4 = *(const v4f*)(crow);
        const v4f u4 = *(const v4f*)(crow - PITCH);
        const v4f d4 = *(const v4f*)(crow + PITCH);
        const float lftv = crow[-1];
        const float rgtv = crow[4];
        const float upM = (gy > 0)      ? 1.0f : 0.0f;   // zero-padding masks
        const float dnM = (gy < NY - 1) ? 1.0f : 0.0f;

        const size_t idx = planeBase + (size_t)gy * NX + gx;
        // Streamed once -> non-temporal: keep L2 for b_geom (8x reuse) and
        // h1 halo rows (shared between vertically adjacent tiles).
        const v4f h2v = __builtin_nontemporal_load((const v4f*)(h2  + idx));
        const v4f clv = __builtin_nontemporal_load((const v4f*)(cl  + idx));
        const v4f rhv = __builtin_nontemporal_load((const v4f*)(rho + idx));
        const v4f bgv = *(const v4f*)(bg + (size_t)gy * NX + gx);  // temporal: reused per batch

        const float llv[4] = {lftOk ? lftv : 0.0f, c4.x, c4.y, c4.z};
        const float rrv[4] = {c4.y, c4.z, c4.w, rgtOk ? rgtv : 0.0f};

        v4f yv;
#pragma unroll
        for (int j = 0; j < 4; ++j) {
            const float h1v  = c4[j];
            const float lap  = u4[j] * upM + d4[j] * dnM
                             + llv[j] + rrv[j] - 4.0f * h1v;      // h = 1
            const float h1sq = h1v * h1v;
            const float bb   = bgv[j] + rhv[j] / (1.0f + h1sq);   // satdamp b0=uth=1
            const float cw   = clv[j] + 0.01f * rhv[j] * h1sq;    // c_nl = 0.01
            const float inv  = 1.0f / (4.0f + 2.0f * bb);         // 1/(dt^-2 + b dt^-1)
            yv[j] = inv * (8.0f * h1v                              // 2/dt^2 = 8
                           - (4.0f - 2.0f * bb) * h2v[j]
                           + (cw * cw) * lap);
        }

        __builtin_nontemporal_store(yv, (v4f*)(out + idx));        // output y
        __builtin_nontemporal_store(c4, (v4f*)(out + TOT + idx));  // output h1 pass-through
    }
}

extern "C" void kernel_launch(void* const* d_in, const int* in_sizes, int n_in,
                              void* d_out, int out_size, void* d_ws, size_t ws_size,
                              hipStream_t stream) {
    (void)in_sizes; (void)n_in; (void)d_ws; (void)ws_size; (void)out_size;
    const float* h1  = (const float*)d_in[0];
    const float* h2  = (const float*)d_in[1];
    const float* cl  = (const float*)d_in[2];
    const float* rho = (const float*)d_in[3];
    const float* bg  = (const float*)d_in[4];
    float* out = (float*)d_out;

    dim3 grid(NX / TILE_X, NY / TILE_Y, BATCH);   // (8, 32, 8)
    dim3 block(256);
    wave_fdtd_kernel<<<grid, block, 0, stream>>>(h1, h2, cl, rho, bg, out);
}